// LastLayer_45019847197065
// MI455X (gfx1250) — compile-verified
//
#include <hip/hip_runtime.h>
#include <hip/hip_bf16.h>

// ---- problem constants (from reference) ----
#define BB     2
#define NN     4096
#define C0     128
#define N0V    10242
#define IN_CH  19
#define K0     147      // IN_CH + C0
#define K0P    148      // padded to multiple of 4 for WMMA K-steps
#define MROWS  8192     // B*N
#define EPSF   1e-05f

typedef __attribute__((ext_vector_type(2))) float v2f;
typedef __attribute__((ext_vector_type(8))) float v8f;

// ---------------------------------------------------------------------------
// Kernel 1: rhs[m] = 0.5f * (r^2 + |v_m|^2 - t^2), where
//   r^2 = |vertex[0]|^2,  t^2 = |vertex[0] - vertex[near_idx[0,0]]|^2
// ---------------------------------------------------------------------------
__global__ void rhs_kernel(const float* __restrict__ vertex,
                           const int* __restrict__ near_idx,
                           float* __restrict__ rhs) {
    int m = blockIdx.x * blockDim.x + threadIdx.x;
    if (m >= N0V) return;
    float v0x = vertex[0], v0y = vertex[1], v0z = vertex[2];
    float r2 = v0x * v0x + v0y * v0y + v0z * v0z;
    int j = near_idx[0];
    float dx = v0x - vertex[3 * j];
    float dy = v0y - vertex[3 * j + 1];
    float dz = v0z - vertex[3 * j + 2];
    float t2 = dx * dx + dy * dy + dz * dz;
    float vx = vertex[3 * m], vy = vertex[3 * m + 1], vz = vertex[3 * m + 2];
    float v2m = vx * vx + vy * vy + vz * vz;
    rhs[m] = 0.5f * (r2 + v2m - t2);
}

// ---------------------------------------------------------------------------
// Kernel 2: copy W0 (128 x 147) into zero-padded W0p (128 x 148)
// ---------------------------------------------------------------------------
__global__ void padw0_kernel(const float* __restrict__ W0,
                             float* __restrict__ W0p) {
    int i = blockIdx.x * blockDim.x + threadIdx.x;
    if (i >= 128 * K0P) return;
    int o = i / K0P, k = i - o * K0P;
    W0p[i] = (k < K0) ? W0[o * K0 + k] : 0.0f;
}

// ---------------------------------------------------------------------------
// Kernel 3: one wave32 per query point. First-hit search over N0 vertices
// (ballot + ctz preserves argmax-of-mask = first-true semantics), then the
// wave cooperatively builds the concatenated row X0[p][0:147] (stride K0P).
// ---------------------------------------------------------------------------
__global__ void feat_kernel(const float* __restrict__ original,   // (B,19,N)
                            const float* __restrict__ projected,  // (B,128,N0)
                            const float* __restrict__ vertex,     // (N0,3)
                            const float* __restrict__ rhs,        // (N0)
                            float* __restrict__ X0) {             // (8192,148)
    int wid  = threadIdx.x >> 5;
    int lane = threadIdx.x & 31;
    int p = blockIdx.x * 8 + wid;          // 1024 blocks * 8 waves = 8192
    int b = p >> 12;                       // p / N
    int n = p & (NN - 1);                  // p % N

    // xyz = orig[b,n,0:3]; pxyz = xyz * r / |xyz|
    float v0x = vertex[0], v0y = vertex[1], v0z = vertex[2];
    float r = sqrtf(v0x * v0x + v0y * v0y + v0z * v0z);
    size_t ob = (size_t)b * IN_CH * NN + n;
    float x = original[ob];
    float y = original[ob + (size_t)NN];
    float z = original[ob + (size_t)2 * NN];
    float s = r * rsqrtf(x * x + y * y + z * z);
    float px = x * s, py = y * s, pz = z * s;

    int first = 0;
    bool found = false;
    for (int base = 0; base < N0V; base += 32) {
        int m = base + lane;
        bool pred = false;
        if (m < N0V) {
            float vx = vertex[3 * m], vy = vertex[3 * m + 1], vz = vertex[3 * m + 2];
            pred = (px * vx + py * vy + pz * vz) >= rhs[m];
        }
        unsigned msk = __builtin_amdgcn_ballot_w32(pred);
        if (msk) { first = base + __builtin_ctz(msk); found = true; break; }
    }

    float* row = X0 + (size_t)p * K0P;
    // original 19 channels
    if (lane < IN_CH) row[lane] = original[ob + (size_t)lane * NN];
    if (lane == IN_CH) row[K0] = 0.0f;   // pad column 147
    // gathered 128-channel feature (or zeros)
    size_t pb = (size_t)b * C0 * N0V + first;
    for (int c = lane; c < C0; c += 32)
        row[IN_CH + c] = found ? projected[pb + (size_t)c * N0V] : 0.0f;
}

// ---------------------------------------------------------------------------
// Kernel 4: GEMM via V_WMMA_F32_16X16X4_F32.  Y(8192x128) = X(8192xK) * W^T + bias
// W is (128 x K) row-major.  Block = 256 thr = 8 waves; block b owns M-tile b,
// wave w owns N-tile w.  EXEC is all-1s (exact tiling, no divergence).
// ---------------------------------------------------------------------------
__global__ void gemm_wmma_kernel(const float* __restrict__ X,
                                 const float* __restrict__ W,
                                 const float* __restrict__ bias,
                                 float* __restrict__ Y,
                                 int K) {
    int wid  = threadIdx.x >> 5;
    int lane = threadIdx.x & 31;
    int half = lane >> 4;          // 0: K pair {0,1}, 1: K pair {2,3}
    int t    = lane & 15;
    int mtile = blockIdx.x;        // 512
    int ntile = wid;               // 8

    const float* arow = X + (size_t)(mtile * 16 + t) * K + 2 * half; // A[t][2h+j]
    const float* brow = W + (size_t)(ntile * 16 + t) * K + 2 * half; // B[2h+j][t]

    v8f acc = {};
    for (int k = 0; k < K; k += 4) {
        v2f a, bv;
        a[0]  = arow[k];     a[1]  = arow[k + 1];
        bv[0] = brow[k];     bv[1] = brow[k + 1];
        acc = __builtin_amdgcn_wmma_f32_16x16x4_f32(
            false, a, false, bv, (short)0, acc, false, false);
    }

    float bz = bias[ntile * 16 + t];
    size_t out = (size_t)(mtile * 16) * 128 + ntile * 16 + t;
    #pragma unroll
    for (int i = 0; i < 8; ++i)                       // D[i+8*half][t]
        Y[out + (size_t)(i + 8 * half) * 128] = acc[i] + bz;
}

// ---------------------------------------------------------------------------
// Kernel 5: per-channel BN stats (deterministic tree reduce, 1 block/channel)
// ab[c] = g*rstd ; ab[128+c] = be - mean*g*rstd
// ---------------------------------------------------------------------------
__global__ void bn_stats_kernel(const float* __restrict__ Y,
                                const float* __restrict__ g,
                                const float* __restrict__ be,
                                float* __restrict__ ab) {
    int c = blockIdx.x;
    float s = 0.0f, ss = 0.0f;
    for (int r = threadIdx.x; r < MROWS; r += 256) {
        float v = Y[(size_t)r * 128 + c];
        s += v; ss += v * v;
    }
    __shared__ float sh0[256], sh1[256];
    sh0[threadIdx.x] = s; sh1[threadIdx.x] = ss;
    __syncthreads();
    for (int off = 128; off > 0; off >>= 1) {
        if (threadIdx.x < off) {
            sh0[threadIdx.x] += sh0[threadIdx.x + off];
            sh1[threadIdx.x] += sh1[threadIdx.x + off];
        }
        __syncthreads();
    }
    if (threadIdx.x == 0) {
        float mean = sh0[0] * (1.0f / MROWS);
        float var  = sh1[0] * (1.0f / MROWS) - mean * mean;
        float a = g[c] * rsqrtf(var + EPSF);
        ab[c] = a;
        ab[128 + c] = be[c] - mean * a;
    }
}

// ---------------------------------------------------------------------------
// Kernel 6: BN apply + ReLU -> next layer input (row-major 8192x128)
// ---------------------------------------------------------------------------
__global__ void bn_apply_kernel(const float* __restrict__ Y,
                                const float* __restrict__ ab,
                                float* __restrict__ Xn) {
    int i = blockIdx.x * blockDim.x + threadIdx.x;   // exact: 8192*128 threads
    int c = i & 127;
    float v = Y[i] * ab[c] + ab[128 + c];
    Xn[i] = v > 0.0f ? v : 0.0f;
}

// ---------------------------------------------------------------------------
// Kernel 7: final BN apply + ReLU, fused with transpose to (B,128,N)
// ---------------------------------------------------------------------------
__global__ void bn_apply_out_kernel(const float* __restrict__ Y,
                                    const float* __restrict__ ab,
                                    float* __restrict__ out) {
    int i = blockIdx.x * blockDim.x + threadIdx.x;
    int c = i & 127;
    int row = i >> 7;
    int b = row >> 12, n = row & (NN - 1);
    float v = Y[i] * ab[c] + ab[128 + c];
    out[((size_t)b * 128 + c) * NN + n] = v > 0.0f ? v : 0.0f;
}

// ---------------------------------------------------------------------------
extern "C" void kernel_launch(void* const* d_in, const int* in_sizes, int n_in,
                              void* d_out, int out_size, void* d_ws, size_t ws_size,
                              hipStream_t stream) {
    const float* original  = (const float*)d_in[0];
    const float* projected = (const float*)d_in[1];
    const float* vertex    = (const float*)d_in[2];
    const int*   near_idx  = (const int*)  d_in[3];
    const float* W0 = (const float*)d_in[4];
    const float* b0 = (const float*)d_in[5];
    const float* g0 = (const float*)d_in[6];
    const float* be0 = (const float*)d_in[7];
    const float* W1 = (const float*)d_in[8];
    const float* b1 = (const float*)d_in[9];
    const float* g1 = (const float*)d_in[10];
    const float* be1 = (const float*)d_in[11];
    const float* W2 = (const float*)d_in[12];
    const float* b2 = (const float*)d_in[13];
    const float* g2 = (const float*)d_in[14];
    const float* be2 = (const float*)d_in[15];
    float* out = (float*)d_out;

    // workspace layout (floats)
    float* ws   = (float*)d_ws;
    float* rhs  = ws;                               // 10242
    float* W0p  = rhs + 10496;                      // 128*148 = 18944
    float* bufA = W0p + 18944;                      // 8192*148 = 1212416
    float* bufB = bufA + (size_t)MROWS * K0P;       // 8192*128 = 1048576
    float* ab   = bufB + (size_t)MROWS * 128;       // 256

    // stage 0: precompute + feature gather
    rhs_kernel<<<(N0V + 255) / 256, 256, 0, stream>>>(vertex, near_idx, rhs);
    padw0_kernel<<<(128 * K0P + 255) / 256, 256, 0, stream>>>(W0, W0p);
    feat_kernel<<<MROWS / 8, 256, 0, stream>>>(original, projected, vertex, rhs, bufA);

    // layer 0: (8192x148)*(148x128)
    gemm_wmma_kernel<<<MROWS / 16, 256, 0, stream>>>(bufA, W0p, b0, bufB, K0P);
    bn_stats_kernel<<<128, 256, 0, stream>>>(bufB, g0, be0, ab);
    bn_apply_kernel<<<MROWS * 128 / 256, 256, 0, stream>>>(bufB, ab, bufA);

    // layer 1
    gemm_wmma_kernel<<<MROWS / 16, 256, 0, stream>>>(bufA, W1, b1, bufB, 128);
    bn_stats_kernel<<<128, 256, 0, stream>>>(bufB, g1, be1, ab);
    bn_apply_kernel<<<MROWS * 128 / 256, 256, 0, stream>>>(bufB, ab, bufA);

    // layer 2 (+ fused transpose to output)
    gemm_wmma_kernel<<<MROWS / 16, 256, 0, stream>>>(bufA, W2, b2, bufB, 128);
    bn_stats_kernel<<<128, 256, 0, stream>>>(bufB, g2, be2, ab);
    bn_apply_out_kernel<<<MROWS * 128 / 256, 256, 0, stream>>>(bufB, ab, out);
}